// Fill2d_47064251629931
// MI455X (gfx1250) — compile-verified
//
#include <hip/hip_runtime.h>

// ---------------------------------------------------------------------------
// Lidar range-image fill pipeline, fixed shapes from the reference:
//   x [64, 2, 128, 2048] f32   (idx_valid is ignored: x>0 <=> idx_valid)
// Two passes, workgroup-per-row, TDM async tile staging into LDS.
//   pass1: y = maxpool3x3_circ(x); y = x>0 ? x : y; fill = mu+sigma per row;
//          y = y>0 ? y : fill   -> workspace
//   pass2: z = y + relu(lap5(y)); out = avgpool5x5_circ(z); out = x>0 ? x : out
// Roofline: ~268 MB min HBM traffic @ 23.3 TB/s => ~11.5 us; memory-bound,
// so fp32 VALU + TDM/LDS data movement is the right CDNA5 path (no GEMM =>
// no WMMA). The 128 MiB y intermediate is L2-resident (192 MB L2).
// ---------------------------------------------------------------------------

#define NN    64
#define CC    2
#define HH    128
#define WW    2048
#define NPL   (NN * CC)        // 128 independent (n,c) planes
#define PLANE (HH * WW)        // 262144 elems per plane
#define HMASK (HH - 1)
#define WMASK (WW - 1)
#define TPB   256
#define CPT   (WW / TPB)       // 8 columns per thread

#if __has_builtin(__builtin_amdgcn_tensor_load_to_lds)
#define USE_TDM 1
#else
#define USE_TDM 0
#endif
#if __has_builtin(__builtin_amdgcn_global_load_async_to_lds_b128)
#define USE_ASYNC 1
#else
#define USE_ASYNC 0
#endif

#if USE_TDM
typedef unsigned int tdm_v4u __attribute__((ext_vector_type(4)));
typedef int          tdm_v8i __attribute__((ext_vector_type(8)));
typedef int          tdm_v4i __attribute__((ext_vector_type(4)));

// One TDM descriptor: copy nrows x 2048 f32 tile global -> LDS (rows packed
// contiguously in LDS, matching float buf[nrows][WW]).
// D# packing per CDNA5 ISA 8.3/8.4 (group0: count/lds/global/type,
// group1: data_size=4B, tensor dims, tile dims, dim0 stride).
// clang-23 6-arg builtin form: (v4u g0, v8i g1, v4i, v4i, v8i, i32 cpol).
__device__ __forceinline__ void tdm_load_rows(const float* g, float* l, int nrows) {
  unsigned long long ga = (unsigned long long)g;
  unsigned int       la = (unsigned int)(unsigned long long)l; // LDS byte offset
  tdm_v4u g0;
  g0[0] = 1u;                                                   // count=1, user mode
  g0[1] = la;                                                   // lds_addr
  g0[2] = (unsigned int)ga;                                     // global_addr[31:0]
  g0[3] = ((unsigned int)(ga >> 32) & 0x01FFFFFFu) | 0x80000000u; // [56:32] | type=2
  tdm_v8i g1;
  g1[0] = (int)(2u << 16);                                      // data_size = 4 bytes
  g1[1] = (int)(((unsigned)WW & 0xFFFFu) << 16);                // tensor_dim0 lo16
  g1[2] = (int)((unsigned)nrows << 16);                         // dim0 hi16=0 | tensor_dim1
  g1[3] = (int)(((unsigned)WW & 0xFFFFu) << 16);                // dim1 hi16=0 | tile_dim0
  g1[4] = nrows;                                                // tile_dim1, tile_dim2=0
  g1[5] = WW;                                                   // tensor_dim0_stride lo32
  g1[6] = 0;                                                    // stride hi | dim1_stride lo
  g1[7] = 0;
  tdm_v4i z4 = {0, 0, 0, 0};
  tdm_v8i z8 = {0, 0, 0, 0, 0, 0, 0, 0};
  __builtin_amdgcn_tensor_load_to_lds(g0, g1, z4, z4, z8, 0);
}
#endif

// Stage one full row (2048 f32) into LDS (generic fallback path and wrap rows).
__device__ __forceinline__ void stage_row(const float* __restrict__ g, float* l) {
#if USE_TDM
  if (threadIdx.x < 32) tdm_load_rows(g, l, 1);
#elif USE_ASYNC
  typedef const __attribute__((address_space(1))) void* gas_t;
  typedef __attribute__((address_space(3))) void*       las_t;
  #pragma unroll
  for (int i = threadIdx.x; i < WW / 4; i += TPB) {
    const float* gp = g + i * 4;
    float*       lp = l + i * 4;
    __builtin_amdgcn_global_load_async_to_lds_b128(
        (gas_t)(unsigned long long)gp,
        (las_t)(unsigned int)(unsigned long long)lp, 0, 0);
  }
#else
  for (int i = threadIdx.x; i < WW / 4; i += TPB)
    ((float4*)l)[i] = ((const float4*)g)[i];
#endif
}

// Stage NR rows starting at plane-row r0 (with circular wrap) into buf.
// Interior blocks use one 2D TDM tile; wrap blocks fall back to row loads.
template <int NR>
__device__ __forceinline__ void stage_rows_wrap(const float* __restrict__ plane,
                                                int r0, float* buf) {
#if USE_TDM
  if (r0 >= 0 && r0 + NR <= HH) {          // contiguous: one 2D descriptor
    if (threadIdx.x < 32) tdm_load_rows(plane + (size_t)r0 * WW, buf, NR);
    return;
  }
#endif
  #pragma unroll
  for (int t = 0; t < NR; ++t)
    stage_row(plane + (size_t)((r0 + t + HH) & HMASK) * WW, buf + t * WW);
}

__device__ __forceinline__ void stage_commit() {
#if USE_TDM
  if (threadIdx.x < 32) __builtin_amdgcn_s_wait_tensorcnt(0);
#elif USE_ASYNC
  #if __has_builtin(__builtin_amdgcn_s_wait_asynccnt)
  __builtin_amdgcn_s_wait_asynccnt(0);
  #else
  asm volatile("s_wait_asynccnt 0" ::: "memory");
  #endif
#endif
  __syncthreads();
}

// Block-wide sum of two values (wave32 shuffle + LDS combine). All 256
// threads end up holding the totals. Leading barrier protects `red` reuse.
__device__ __forceinline__ void block_sum2(float& a, float& b, float* red) {
  #pragma unroll
  for (int off = 16; off > 0; off >>= 1) {
    a += __shfl_xor(a, off, 32);
    b += __shfl_xor(b, off, 32);
  }
  __syncthreads();
  if ((threadIdx.x & 31) == 0) {
    red[threadIdx.x >> 5]       = a;
    red[8 + (threadIdx.x >> 5)] = b;
  }
  __syncthreads();
  float ta = 0.f, tb = 0.f;
  #pragma unroll
  for (int i = 0; i < TPB / 32; ++i) { ta += red[i]; tb += red[8 + i]; }
  a = ta; b = tb;
}

// ---------------------------------------------------------------------------
// Pass 1: circular 3x3 maxpool + valid restore + per-row (mu + sigma) fill.
// One block per (plane,row); y written to workspace (chunk-local layout).
// ---------------------------------------------------------------------------
__global__ __launch_bounds__(TPB) void pass1_maxpool_fill(
    const float* __restrict__ x, float* __restrict__ y, int planeBase) {
  __shared__ __align__(16) float xb[3][WW];   // 24 KB: rows r-1, r, r+1
  __shared__ float red[16];

  const int tid = threadIdx.x;
  const int r   = blockIdx.x & HMASK;
  const int p   = planeBase + (blockIdx.x >> 7);
  const float* xp   = x + (size_t)p * PLANE;
  float*       yrow = y + (size_t)blockIdx.x * WW;   // chunk-local ws row

  stage_rows_wrap<3>(xp, r - 1, &xb[0][0]);
  stage_commit();

  float yv[CPT];
  float cnt = 0.f, sum = 0.f;
  #pragma unroll
  for (int k = 0; k < CPT; ++k) {
    const int j  = tid + k * TPB;
    const int jm = (j - 1) & WMASK, jp = (j + 1) & WMASK;
    float m0 = fmaxf(fmaxf(xb[0][jm], xb[0][j]), xb[0][jp]);
    float m1 = fmaxf(fmaxf(xb[1][jm], xb[1][j]), xb[1][jp]);
    float m2 = fmaxf(fmaxf(xb[2][jm], xb[2][j]), xb[2][jp]);
    float m  = fmaxf(fmaxf(m0, m1), m2);
    float xc = xb[1][j];
    float v  = (xc > 0.f) ? xc : m;      // where(idx_valid, x, maxpool)
    yv[k] = v;
    if (v > 0.f) { cnt += 1.f; sum += v; }
  }

  block_sum2(cnt, sum, red);
  const float mu = sum / cnt;

  // Two-pass variance, matching reference sum((y-mu)^2 * idx)/cnt.
  float s2 = 0.f, dummy = 0.f;
  #pragma unroll
  for (int k = 0; k < CPT; ++k) {
    if (yv[k] > 0.f) { float d = yv[k] - mu; s2 += d * d; }
  }
  block_sum2(s2, dummy, red);
  const float fill = mu + sqrtf(s2 / cnt);

  #pragma unroll
  for (int k = 0; k < CPT; ++k) {
    float v = yv[k];
    yrow[tid + k * TPB] = (v > 0.f) ? v : fill;
  }
}

// ---------------------------------------------------------------------------
// Pass 2: z = y + relu(lap(y)); out = avg5x5_circ(z); out = x>0 ? x : out.
// One block per output row; needs y rows r-3..r+3 (y is L2-resident).
// LDS: 56 KB y + 40 KB z + 8 KB column sums = 104 KB (3 blocks / 320 KB WGP).
// ---------------------------------------------------------------------------
__global__ __launch_bounds__(TPB) void pass2_lap_avg(
    const float* __restrict__ x, const float* __restrict__ y,
    float* __restrict__ out, int planeBase) {
  __shared__ __align__(16) float yb[7][WW];   // 56 KB
  __shared__ __align__(16) float zb[5][WW];   // 40 KB
  __shared__ __align__(16) float vb[WW];      //  8 KB vertical sums

  const int tid = threadIdx.x;
  const int r   = blockIdx.x & HMASK;
  const int p   = planeBase + (blockIdx.x >> 7);
  const float* yp = y + (size_t)(blockIdx.x >> 7) * PLANE;  // chunk-local plane

  stage_rows_wrap<7>(yp, r - 3, &yb[0][0]);

  // Prefetch the x row needed for the final where(); one prefetch per 32 B
  // stride covers every cacheline of the 8 KB row (global_prefetch_b8).
  const float* xrow = x + (size_t)p * PLANE + (size_t)r * WW;
  __builtin_prefetch(xrow + tid * CPT, 0, 3);

  stage_commit();

  #pragma unroll
  for (int u = 0; u < 5; ++u) {               // z row (r-2+u), center yb[u+1]
    #pragma unroll
    for (int k = 0; k < CPT; ++k) {
      const int j  = tid + k * TPB;
      const int jm = (j - 1) & WMASK, jp = (j + 1) & WMASK;
      float c   = yb[u + 1][j];
      float lap = yb[u][j] + yb[u + 2][j] + yb[u + 1][jm] + yb[u + 1][jp] - 4.f * c;
      zb[u][j]  = c + fmaxf(lap, 0.f);        // y + relu(laplacian)
    }
  }
  __syncthreads();

  #pragma unroll
  for (int k = 0; k < CPT; ++k) {
    const int j = tid + k * TPB;
    vb[j] = zb[0][j] + zb[1][j] + zb[2][j] + zb[3][j] + zb[4][j];
  }
  __syncthreads();

  float* orow = out + (size_t)p * PLANE + (size_t)r * WW;
  #pragma unroll
  for (int k = 0; k < CPT; ++k) {
    const int j = tid + k * TPB;
    float s = vb[(j - 2) & WMASK] + vb[(j - 1) & WMASK] + vb[j] +
              vb[(j + 1) & WMASK] + vb[(j + 2) & WMASK];
    float xc = xrow[j];
    orow[j] = (xc > 0.f) ? xc : s * (1.f / 25.f);   // final where(idx_valid,...)
  }
}

// ---------------------------------------------------------------------------
// Host: chunk over (n,c) planes so any ws_size >= 1 MiB works; with
// ws_size >= 128 MiB this is exactly two kernel launches.
// ---------------------------------------------------------------------------
extern "C" void kernel_launch(void* const* d_in, const int* in_sizes, int n_in,
                              void* d_out, int out_size, void* d_ws, size_t ws_size,
                              hipStream_t stream) {
  const float* x   = (const float*)d_in[0];   // idx_valid (d_in[1]) unused: x>0 <=> valid
  float*       out = (float*)d_out;
  float*       ws  = (float*)d_ws;

  const size_t planeBytes = (size_t)PLANE * sizeof(float);
  int chunk = (int)(ws_size / planeBytes);
  if (chunk < 1)   chunk = 1;
  if (chunk > NPL) chunk = NPL;

  for (int base = 0; base < NPL; base += chunk) {
    const int np = (NPL - base < chunk) ? (NPL - base) : chunk;
    dim3 grid((unsigned)(np * HH));
    pass1_maxpool_fill<<<grid, TPB, 0, stream>>>(x, ws, base);
    pass2_lap_avg<<<grid, TPB, 0, stream>>>(x, ws, out, base);
  }
  (void)in_sizes; (void)n_in; (void)out_size;
}